// AddSpatialInfo_5076651344310
// MI455X (gfx1250) — compile-verified
//
#include <hip/hip_runtime.h>
#include <cstdint>

// ---------------------------------------------------------------------------
// AddSpatialInfo for MI455X (gfx1250):
//   out[b, 0:2048, i, j] = in[b, :, i, j]
//   out[b, 2048,  i, j]  = j*2/W - 1
//   out[b, 2049,  i, j]  = i*2/H - 1
// Pure data movement (0 FLOPs): WMMA is inapplicable; the CDNA5-specific
// path here is the ASYNCcnt-tracked global<->LDS DMA (bypasses the VGPR
// file) plus temporal-hint cache control.
// Roofline: ~103 MB read + ~103 MB write. With RT loads the input stays
// resident in the 192 MB L2 across graph replays; NT stores keep the output
// stream from evicting it -> replays are HBM-write-bound (~4.4 us floor).
// ---------------------------------------------------------------------------

typedef float v4f __attribute__((ext_vector_type(4)));
typedef int   v4i __attribute__((vector_size(16)));   // matches builtin param

#define BATCH      64
#define CHAN       2048
#define HH         14
#define WW         14
#define PLANE      (HH * WW)             /* 196 floats per plane           */
#define PLANE4     (PLANE / 4)           /* 49 vec4 per plane              */
#define IN_BATCH4  (CHAN * PLANE4)       /* 100352 vec4 per input batch    */
#define OUT_BATCH4 ((CHAN + 2) * PLANE4) /* 100450 vec4 per output batch   */
#define TOTAL4     (BATCH * OUT_BATCH4)  /* 6,428,800 output vec4          */
#define COORD4     (BATCH * 2 * PLANE4)  /* 6,272 coordinate vec4          */

#define THREADS    256
#define VPT        8                      /* vec4 per thread               */
#define TILE4      (THREADS * VPT)        /* 2048 vec4 (32 KB) per block   */
#define BLKS_PER_B (IN_BATCH4 / TILE4)    /* 100352 / 2048 = 49, exact     */

// Detect gfx1250 async global<->LDS builtins (confirmed present: round 2
// compiled and lowered them).
#if defined(__has_builtin)
#  if __has_builtin(__builtin_amdgcn_global_load_async_to_lds_b128) && \
      __has_builtin(__builtin_amdgcn_global_store_async_from_lds_b128)
#    define HAVE_ASYNC_LDS 1
#  endif
#endif
#ifndef HAVE_ASYNC_LDS
#  define HAVE_ASYNC_LDS 0
#endif

// Builtin params: NON-const pointer to a 16-byte int vector in the
// global (1) / LDS (3) address spaces (shape confirmed by round-1 diag).
#define AS1_V4I(p) ((__attribute__((address_space(1))) v4i*)(p))
#define AS3_V4I(p) ((__attribute__((address_space(3))) v4i*)(p))

// gfx12 CPol immediate: TH in bits [2:0] (TH_STORE_NT == 1), scope [4:3].
#define CPOL_RT       0
#define CPOL_STORE_NT 1

__device__ __forceinline__ void wait_async_zero() {
#if defined(__has_builtin) && __has_builtin(__builtin_amdgcn_s_wait_asynccnt)
    __builtin_amdgcn_s_wait_asynccnt(0);
#else
    asm volatile("s_wait_asynccnt 0x0" ::: "memory");
#endif
}

__device__ __forceinline__ v4f coord_vec(uint32_t ch, uint32_t q) {
    // ch 0 -> x channel (j-based), ch 1 -> y channel (i-based)
    v4f val;
#pragma unroll
    for (int k = 0; k < 4; ++k) {
        uint32_t p = q * 4u + (uint32_t)k;   // 0..195 within plane
        uint32_t i = p / WW;
        uint32_t j = p - i * WW;
        uint32_t t = (ch == 0u) ? j : i;
        // Match jnp f32 order: (t * 2.0f) / dim - 1.0f
        float dim = (ch == 0u) ? (float)WW : (float)HH;
        val[k] = ((float)t * 2.0f) / dim - 1.0f;
    }
    return val;
}

#if HAVE_ASYNC_LDS
// ---------------------------------------------------------------------------
// Bulk copy via CDNA5 async DMA. Grid = (49, 64): blockIdx.y = batch,
// blockIdx.x = 2048-vec4 tile within the batch. No divisions, no tails.
// Each thread: 8 async loads into private LDS slots (4 KB of ASYNCcnt
// traffic in flight per wave), one s_wait_asynccnt 0, 8 async NT stores.
// Per-thread slots -> no cross-thread deps -> no workgroup barrier needed
// (ASYNCcnt is per-wave). S_ENDPGM's implicit wait-idle drains the stores.
// ---------------------------------------------------------------------------
__global__ __launch_bounds__(THREADS) void bulk_copy_async(
        const float* __restrict__ in, float* __restrict__ out) {
    __shared__ __align__(16) v4f stage[TILE4];        // 32 KB
    uint32_t b    = blockIdx.y;
    uint32_t base = blockIdx.x * (uint32_t)TILE4;     // vec4 index in batch
    const v4f* gsrc = (const v4f*)in  + (size_t)b * IN_BATCH4  + base;
    v4f*       gdst = (v4f*)out       + (size_t)b * OUT_BATCH4 + base;
    uint32_t t = threadIdx.x;

#pragma unroll
    for (int k = 0; k < VPT; ++k) {
        uint32_t idx = (uint32_t)k * THREADS + t;     // coalesced 512B/wave
        __builtin_amdgcn_global_load_async_to_lds_b128(
            AS1_V4I(gsrc + idx), AS3_V4I(&stage[idx]),
            /*offset=*/0, /*cpol=*/CPOL_RT);
    }

    wait_async_zero();

#pragma unroll
    for (int k = 0; k < VPT; ++k) {
        uint32_t idx = (uint32_t)k * THREADS + t;
        __builtin_amdgcn_global_store_async_from_lds_b128(
            AS1_V4I(gdst + idx), AS3_V4I(&stage[idx]),
            /*offset=*/0, /*cpol=*/CPOL_STORE_NT);
    }
}
#else
// ---------------------------------------------------------------------------
// Portability fallback (not compiled on this toolchain): fused vec4 copy.
// ---------------------------------------------------------------------------
__global__ __launch_bounds__(THREADS) void fused_concat(
        const v4f* __restrict__ in4, v4f* __restrict__ out4) {
    uint32_t v = blockIdx.x * (uint32_t)THREADS + threadIdx.x;
    if (v >= TOTAL4) return;
    uint32_t b = v / OUT_BATCH4;
    uint32_t r = v - b * OUT_BATCH4;
    v4f val;
    if (r < IN_BATCH4) {
        val = in4[b * IN_BATCH4 + r];
    } else {
        uint32_t p4 = r - IN_BATCH4;
        uint32_t ch = p4 / PLANE4;
        uint32_t q  = p4 - ch * PLANE4;
        val = coord_vec(ch, q);
    }
    __builtin_nontemporal_store(val, out4 + v);
}
#endif  // HAVE_ASYNC_LDS

// Coordinate channels: 6,272 vec4 total -> negligible; plain NT stores.
__global__ __launch_bounds__(THREADS) void coord_fill(float* __restrict__ out) {
    uint32_t idx = blockIdx.x * (uint32_t)THREADS + threadIdx.x;
    if (idx >= COORD4) return;
    uint32_t b  = idx / (2u * PLANE4);
    uint32_t r  = idx - b * (2u * PLANE4);
    uint32_t ch = r / PLANE4;
    uint32_t q  = r - ch * PLANE4;
    v4f val = coord_vec(ch, q);
    v4f* dst = (v4f*)out +
               ((size_t)b * OUT_BATCH4 + IN_BATCH4 + ch * PLANE4 + q);
    __builtin_nontemporal_store(val, dst);
}

extern "C" void kernel_launch(void* const* d_in, const int* in_sizes, int n_in,
                              void* d_out, int out_size, void* d_ws, size_t ws_size,
                              hipStream_t stream) {
    const float* in = (const float*)d_in[0];
    float* out = (float*)d_out;
    (void)in_sizes; (void)n_in; (void)out_size; (void)d_ws; (void)ws_size;

#if HAVE_ASYNC_LDS
    dim3 grid(BLKS_PER_B, BATCH);                     // (49, 64)
    bulk_copy_async<<<grid, THREADS, 0, stream>>>(in, out);
#else
    fused_concat<<<(TOTAL4 + THREADS - 1) / THREADS, THREADS, 0, stream>>>(
        (const v4f*)in, (v4f*)out);
#endif
    coord_fill<<<(COORD4 + THREADS - 1) / THREADS, THREADS, 0, stream>>>(out);
}